// SheafConvLayer_47115791237987
// MI455X (gfx1250) — compile-verified
//
#include <hip/hip_runtime.h>
#include <math.h>

#define DD 32
#define STEPF 1.0f

typedef __attribute__((ext_vector_type(2))) float v2f;
typedef __attribute__((ext_vector_type(8))) float v8f;

__global__ void zero_f32_kernel(float* __restrict__ p, int n) {
    int i = blockIdx.x * blockDim.x + threadIdx.x;
    if (i < n) p[i] = 0.0f;
}

// One wave (32 lanes) per edge. Lane d handles feature d of both endpoint rows
// (fully coalesced 128B row reads). Shuffle-reduce the 2D-length dot product,
// tanh on lane 0, atomic accumulate maps^2 into diag_maps[row].
__global__ void edge_maps_kernel(const float* __restrict__ x,
                                 const float* __restrict__ sheaf_W,
                                 const int* __restrict__ edge_index,
                                 float* __restrict__ maps,
                                 float* __restrict__ diag_maps,
                                 int E) {
    int wave = (blockIdx.x * blockDim.x + threadIdx.x) >> 5;
    int lane = threadIdx.x & 31;
    if (wave >= E) return;
    int r = edge_index[wave];
    int c = edge_index[E + wave];
    float p = sheaf_W[lane]      * x[r * DD + lane]
            + sheaf_W[DD + lane] * x[c * DD + lane];
    #pragma unroll
    for (int off = 16; off > 0; off >>= 1)
        p += __shfl_xor(p, off, 32);
    if (lane == 0) {
        float m = tanhf(p);
        maps[wave] = m;
        atomicAdd(&diag_maps[r], m * m);
    }
}

__global__ void node_diag_kernel(const float* __restrict__ diag_maps,
                                 float* __restrict__ d_si,
                                 float* __restrict__ diag,
                                 int N) {
    int i = blockIdx.x * blockDim.x + threadIdx.x;
    if (i >= N) return;
    float v = diag_maps[i];
    float s = rsqrtf(v + 1.0f);
    d_si[i] = s;
    diag[i] = s * v * s;
}

// Y = X @ lin_W^T + b via V_WMMA_F32_16X16X4_F32.
// One wave -> one 16x16 tile of Y (tileM in rows, tileN in {0,1} for 32 cols).
// A[m][k] = X[tileM*16+m][k], B[k][n] = lin_W[tileN*16+n][k] (so B loads are
// contiguous float2 from lin_W rows). C preloaded with broadcast bias.
// Fused epilogue: y store + out = x - STEP*diag*y.
__global__ void gemm_wmma_kernel(const float* __restrict__ x,
                                 const float* __restrict__ lin_W,
                                 const float* __restrict__ lin_b,
                                 const float* __restrict__ diag,
                                 float* __restrict__ y,
                                 float* __restrict__ out,
                                 int N, int nTilesM) {
    int wave = (blockIdx.x * blockDim.x + threadIdx.x) >> 5;
    int lane = threadIdx.x & 31;
    if (wave >= nTilesM * 2) return;          // whole-wave exit: EXEC all-1s for WMMA
    int tileM = wave >> 1;
    int tileN = wave & 1;
    int g  = lane >> 4;                        // half-wave select (K pairing)
    int ln = lane & 15;                        // M index for A, N index for B/C/D
    int rowA = tileM * 16 + ln;
    if (rowA >= N) rowA = N - 1;               // defensive clamp (N%16==0 here)
    int colN = tileN * 16 + ln;

    v8f c;
    float bias = lin_b[colN];
    #pragma unroll
    for (int rr = 0; rr < 8; ++rr) c[rr] = bias;

    const float* xrow = x     + (size_t)rowA * DD;
    const float* wrow = lin_W + (size_t)colN * DD;
    #pragma unroll
    for (int kk = 0; kk < 8; ++kk) {
        int k = kk * 4 + 2 * g;
        v2f a = *(const v2f*)(xrow + k);       // A[m][k], A[m][k+1]
        v2f b = *(const v2f*)(wrow + k);       // B[k][n], B[k+1][n] = lin_W[n][k..k+1]
        c = __builtin_amdgcn_wmma_f32_16x16x4_f32(
                /*neg_a=*/false, a, /*neg_b=*/false, b,
                /*c_mod=*/(short)0, c, /*reuse_a=*/false, /*reuse_b=*/false);
    }

    // D layout: VGPR rr, lane l -> M = rr + 8*(l/16), N = l%16
    #pragma unroll
    for (int rr = 0; rr < 8; ++rr) {
        int m = tileM * 16 + rr + 8 * g;
        if (m < N) {
            size_t idx = (size_t)m * DD + colN;
            float yv = c[rr];
            y[idx] = yv;
            out[idx] = x[idx] - STEPF * diag[m] * yv;
        }
    }
}

// One wave per edge: compute norm_maps on the fly (scalar loads hit cache),
// lane d gathers y[col][d] (coalesced 128B) and atomically adds -STEP*nm*y
// into out[row][d] (GLOBAL_ATOMIC_ADD_F32, L2 atomic units).
__global__ void edge_scatter_kernel(const float* __restrict__ maps,
                                    const float* __restrict__ d_si,
                                    const float* __restrict__ y,
                                    const int* __restrict__ edge_index,
                                    const int* __restrict__ right_idx,
                                    float* __restrict__ out,
                                    int E) {
    int wave = (blockIdx.x * blockDim.x + threadIdx.x) >> 5;
    int lane = threadIdx.x & 31;
    if (wave >= E) return;
    int r = edge_index[wave];
    int c = edge_index[E + wave];
    float nm = -(maps[wave] * maps[right_idx[wave]]) * d_si[r] * d_si[c];
    float contrib = -STEPF * nm * y[(size_t)c * DD + lane];
    atomicAdd(&out[(size_t)r * DD + lane], contrib);
}

extern "C" void kernel_launch(void* const* d_in, const int* in_sizes, int n_in,
                              void* d_out, int out_size, void* d_ws, size_t ws_size,
                              hipStream_t stream) {
    const float* x       = (const float*)d_in[0];
    const float* sheaf_W = (const float*)d_in[1];
    const float* lin_W   = (const float*)d_in[2];
    const float* lin_b   = (const float*)d_in[3];
    const int*   eidx    = (const int*)d_in[4];
    const int*   ridx    = (const int*)d_in[5];
    float* out = (float*)d_out;

    int N = in_sizes[0] / DD;      // 50000
    int E = in_sizes[4] / 2;       // 1,600,000 directed edges

    // Workspace layout (fp32): maps[E] | diag_maps[N] | d_si[N] | diag[N] | y[N*D]
    float* ws        = (float*)d_ws;
    float* maps      = ws;
    float* diag_maps = maps + E;
    float* d_si      = diag_maps + N;
    float* diag      = d_si + N;
    float* yb        = diag + N;

    zero_f32_kernel<<<(N + 255) / 256, 256, 0, stream>>>(diag_maps, N);

    {   // 8 waves per 256-thread block, one edge per wave
        int blocks = (E + 7) / 8;
        edge_maps_kernel<<<blocks, 256, 0, stream>>>(x, sheaf_W, eidx, maps, diag_maps, E);
    }

    node_diag_kernel<<<(N + 255) / 256, 256, 0, stream>>>(diag_maps, d_si, diag, N);

    {   // WMMA GEMM: nTilesM*2 wave-tasks, 4 waves per 128-thread block
        int nTilesM = (N + 15) / 16;
        int tasks = nTilesM * 2;
        int blocks = (tasks + 3) / 4;
        gemm_wmma_kernel<<<blocks, 128, 0, stream>>>(x, lin_W, lin_b, diag, yb, out, N, nTilesM);
    }

    {   // edge scatter into out
        int blocks = (E + 7) / 8;
        edge_scatter_kernel<<<blocks, 256, 0, stream>>>(maps, d_si, yb, eidx, ridx, out, E);
    }
}